// TextClassificationModel_20753281975116
// MI455X (gfx1250) — compile-verified
//
#include <hip/hip_runtime.h>

typedef __attribute__((ext_vector_type(2))) float v2f;
typedef __attribute__((ext_vector_type(8))) float v8f;

#define N_BAG 16384
#define FEAT 64
#define DOF 64
#define N_CLS 4
#define NPAD 16           // B-matrix N padded to 16 (cols 4..15 zero)
#define BAGS_PER_BLOCK 16
#define THREADS 512       // 16 waves of 32

// One block = 16 bags. Wave w accumulates bag w's normalized-row sum into
// registers, deposits the 64-float pooled vector in LDS, then wave 0 does the
// (16 x 64) @ (64 x 4) projection with V_WMMA_F32_16X16X4_F32 using the
// pre-fused W_c = W_backbone @ W_fc (computed redundantly per block, and
// stored zero-padded to 64x16 so the B operand needs no exec-masked selects).
__global__ __launch_bounds__(THREADS) void bag_fused_kernel(
    const int* __restrict__ text,
    const int* __restrict__ offsets,
    const float* __restrict__ COUNT,
    const float* __restrict__ Wb,     // FEAT x DOF
    const float* __restrict__ Wf,     // DOF x N_CLS
    float* __restrict__ out)          // N_BAG x N_CLS
{
    __shared__ float s_wc[FEAT][NPAD];               // W_c padded: cols 4..15 = 0
    __shared__ float s_pooled[BAGS_PER_BLOCK][FEAT]; // per-bag pooled vectors

    const int tid = threadIdx.x;

    // ---- Phase 0a: zero the padding columns 4..15 (disjoint from 0b) ----
    for (int idx = tid; idx < FEAT * (NPAD - N_CLS); idx += THREADS) {
        const int f = idx / (NPAD - N_CLS);
        const int c = N_CLS + idx % (NPAD - N_CLS);
        s_wc[f][c] = 0.f;
    }

    // ---- Phase 0b: W_c = W_backbone @ W_fc (64x4), threads 0..255 ----
    if (tid < FEAT * N_CLS) {
        const int f = tid >> 2;
        const int c = tid & 3;
        float s = 0.f;
        #pragma unroll 8
        for (int d = 0; d < DOF; ++d)
            s += Wb[f * DOF + d] * Wf[d * N_CLS + c];
        s_wc[f][c] = s;
    }

    // ---- Phase 1: per-wave bag accumulation (L2-gather-bound part) ----
    const int wave  = tid >> 5;
    const int lane  = tid & 31;
    const int sub   = lane & 3;   // which 16-float slice of the 64-float row
    const int tslot = lane >> 2;  // token slot 0..7 (8 tokens in flight)
    const int bag   = blockIdx.x * BAGS_PER_BLOCK + wave;
    const int start = offsets[bag];
    const int end   = offsets[bag + 1];

    float acc[16];
    #pragma unroll
    for (int i = 0; i < 16; ++i) acc[i] = 0.f;

    for (int t = start + tslot; t < end; t += 8) {
        const int tok = text[t];
        const float4* rp =
            (const float4*)(COUNT + (size_t)tok * FEAT + sub * 16);
        const float4 r0 = rp[0];
        const float4 r1 = rp[1];
        const float4 r2 = rp[2];
        const float4 r3 = rp[3];

        // full row sum across the quad of lanes sharing this token
        float p = (r0.x + r0.y + r0.z + r0.w) + (r1.x + r1.y + r1.z + r1.w)
                + (r2.x + r2.y + r2.z + r2.w) + (r3.x + r3.y + r3.z + r3.w);
        p += __shfl_xor(p, 1, 32);
        p += __shfl_xor(p, 2, 32);

        const float denom = fmaxf(p - 1.f, 0.f) + 1.f;   // relu(sum-1)+1
        const float s = 1.f / denom;

        acc[ 0] += r0.x * s; acc[ 1] += r0.y * s; acc[ 2] += r0.z * s; acc[ 3] += r0.w * s;
        acc[ 4] += r1.x * s; acc[ 5] += r1.y * s; acc[ 6] += r1.z * s; acc[ 7] += r1.w * s;
        acc[ 8] += r2.x * s; acc[ 9] += r2.y * s; acc[10] += r2.z * s; acc[11] += r2.w * s;
        acc[12] += r3.x * s; acc[13] += r3.y * s; acc[14] += r3.z * s; acc[15] += r3.w * s;
    }

    // reduce across the 8 token slots (lanes with identical `sub`)
    #pragma unroll
    for (int i = 0; i < 16; ++i) {
        acc[i] += __shfl_xor(acc[i],  4, 32);
        acc[i] += __shfl_xor(acc[i],  8, 32);
        acc[i] += __shfl_xor(acc[i], 16, 32);
    }

    // pooled /= (relu(len-1)+1)
    const float inv_len = 1.f / (fmaxf((float)(end - start - 1), 0.f) + 1.f);
    if (lane < 4) {
        #pragma unroll
        for (int i = 0; i < 16; ++i)
            s_pooled[wave][sub * 16 + i] = acc[i] * inv_len;
    }

    __syncthreads();

    // ---- Phase 2: wave 0 projects pooled(16x64) @ W_c(64x16pad) via WMMA ----
    if (wave == 0) {
        const int row = lane & 15;   // A-matrix M index for this lane
        const int hi  = lane >> 4;   // lanes 16-31 carry K+2 / K+3
        const int col = lane & 15;   // B/C/D N index

        v8f cacc = {0.f, 0.f, 0.f, 0.f, 0.f, 0.f, 0.f, 0.f};

        #pragma unroll
        for (int k = 0; k < 16; ++k) {           // 16 K-chunks of 4 -> K=64
            const int kb = 4 * k + 2 * hi;
            v2f a, b;
            a.x = s_pooled[row][kb];
            a.y = s_pooled[row][kb + 1];
            b.x = s_wc[kb][col];                 // unconditional: pad is zero
            b.y = s_wc[kb + 1][col];
            cacc = __builtin_amdgcn_wmma_f32_16x16x4_f32(
                /*neg_a=*/false, a, /*neg_b=*/false, b,
                /*c_mod=*/(short)0, cacc, /*reuse_a=*/false, /*reuse_b=*/false);
        }

        const int bagbase = blockIdx.x * BAGS_PER_BLOCK;
        #pragma unroll
        for (int j = 0; j < 8; ++j) {
            const int M = 8 * hi + j;            // C/D: VGPR j -> M=j / M=8+j
            if (col < N_CLS)
                out[(bagbase + M) * N_CLS + col] = cacc[j];
        }
    }
}

extern "C" void kernel_launch(void* const* d_in, const int* in_sizes, int n_in,
                              void* d_out, int out_size, void* d_ws, size_t ws_size,
                              hipStream_t stream) {
    (void)in_sizes; (void)n_in; (void)out_size; (void)d_ws; (void)ws_size;
    const int*   text    = (const int*)d_in[0];
    const int*   offsets = (const int*)d_in[1];
    const float* COUNT   = (const float*)d_in[2];
    const float* Wb      = (const float*)d_in[3];
    const float* Wf      = (const float*)d_in[4];
    float*       out     = (float*)d_out;

    bag_fused_kernel<<<N_BAG / BAGS_PER_BLOCK, THREADS, 0, stream>>>(
        text, offsets, COUNT, Wb, Wf, out);
}